// CentralDiff2D_78125455114591
// MI455X (gfx1250) — compile-verified
//
#include <hip/hip_runtime.h>
#include <hip/hip_bf16.h>
#include <stdint.h>

// ---------------------------------------------------------------------------
// CentralDiff2D on MI455X (gfx1250)
//  - grid (H*W floats = 64MB) lives in d_ws, resident in the 192MB L2
//  - zero-fill uses CDNA5 async LDS->global stores (ASYNCcnt path)
//  - point streams (coords/feats/out) use NT hints + 2-points-per-lane
//    vectorization (B128 coord loads) to protect L2 residency of the grid
// ---------------------------------------------------------------------------

typedef int   v4i __attribute__((ext_vector_type(4)));
typedef float v2f __attribute__((ext_vector_type(2)));

// Kernel 1: zero the grid with global_store_async_from_lds_b128.
// Each lane owns 16 bytes of zeros in LDS and issues async 16B stores over a
// grid-stride range (wave covers 512B contiguous per issue).
__global__ void __launch_bounds__(256)
cd2d_zero_kernel(float* __restrict__ grid,
                 const int* __restrict__ pH,
                 const int* __restrict__ pW) {
    __shared__ float lzero[256 * 4];  // 4KB of zeros, 16B per lane

    const int t = threadIdx.x;
    float4 z; z.x = 0.f; z.y = 0.f; z.z = 0.f; z.w = 0.f;
    reinterpret_cast<float4*>(lzero)[t] = z;
    __syncthreads();

    // Low 32 bits of a generic pointer to LDS == raw LDS byte offset.
    const uint32_t lds_off =
        (uint32_t)(uintptr_t)(&reinterpret_cast<float4*>(lzero)[t]);

    const size_t total = (size_t)(*pH) * (size_t)(*pW);   // elements
    const size_t nvec  = total >> 2;                      // float4 chunks
    const size_t stride = (size_t)gridDim.x * (size_t)blockDim.x;
    size_t i = (size_t)blockIdx.x * (size_t)blockDim.x + (size_t)t;

    // Make LDS zeros visible to the async DMA path before issuing.
    asm volatile("s_wait_dscnt 0" ::: "memory");

    const float4* gbase = reinterpret_cast<const float4*>(grid);
    for (; i < nvec; i += stride) {
        const uint64_t gaddr = (uint64_t)(uintptr_t)(gbase + i);
        asm volatile("global_store_async_from_lds_b128 %0, %1, off"
                     :: "v"(gaddr), "v"(lds_off)
                     : "memory");
    }
    asm volatile("s_wait_asynccnt 0" ::: "memory");

    // Scalar tail (H*W not divisible by 4).
    for (size_t j = (nvec << 2) +
                    (size_t)blockIdx.x * (size_t)blockDim.x + (size_t)t;
         j < total; j += stride) {
        grid[j] = 0.f;
    }
}

// Kernel 2: scatter feats into the grid, 2 points per lane.
// coords: one B128 NT load -> {x0,y0,x1,y1}; feats: B64 NT load.
// Grid stores use default (RT) policy -> stay resident in L2.
__global__ void __launch_bounds__(256)
cd2d_scatter_kernel(const v4i* __restrict__ coords2,
                    const v2f* __restrict__ feats2,
                    float* __restrict__ grid,
                    const int* __restrict__ pW,
                    int n) {
    const int i = blockIdx.x * blockDim.x + threadIdx.x;  // pair index
    const int npair = n >> 1;
    const int W = *pW;
    if (i < npair) {
        const v4i c = __builtin_nontemporal_load(&coords2[i]);
        const v2f f = __builtin_nontemporal_load(&feats2[i]);
        grid[(size_t)c.y * (size_t)W + (size_t)c.x] = f.x;
        grid[(size_t)c.w * (size_t)W + (size_t)c.z] = f.y;
    } else if ((n & 1) && i == npair) {
        // odd tail: last point
        const int* cs = (const int*)coords2;
        const int x = cs[2 * (n - 1)], y = cs[2 * (n - 1) + 1];
        grid[(size_t)y * (size_t)W + (size_t)x] =
            ((const float*)feats2)[n - 1];
    }
}

// Kernel 3: gather neighbors from the (L2-resident) grid, 2 points per lane.
// out = 0.5*grid[lin+1]*(x+1<W) - 0.5*grid[lin-1]*(x>0)
__device__ __forceinline__ float cd2d_tap(const float* __restrict__ grid,
                                          int x, int y, int W) {
    const size_t lin = (size_t)y * (size_t)W + (size_t)x;
    float r = 0.f;
    if (x + 1 < W) r  = 0.5f * grid[lin + 1];
    if (x > 0)     r -= 0.5f * grid[lin - 1];
    return r;
}

__global__ void __launch_bounds__(256)
cd2d_gather_kernel(const v4i* __restrict__ coords2,
                   const float* __restrict__ grid,
                   float* __restrict__ out,
                   const int* __restrict__ pW,
                   int n) {
    const int i = blockIdx.x * blockDim.x + threadIdx.x;  // pair index
    const int npair = n >> 1;
    const int W = *pW;
    if (i < npair) {
        const v4i c = __builtin_nontemporal_load(&coords2[i]);
        v2f r;
        r.x = cd2d_tap(grid, c.x, c.y, W);
        r.y = cd2d_tap(grid, c.z, c.w, W);
        __builtin_nontemporal_store(r, (v2f*)&out[2 * (size_t)i]);
    } else if ((n & 1) && i == npair) {
        const int* cs = (const int*)coords2;
        const int x = cs[2 * (n - 1)], y = cs[2 * (n - 1) + 1];
        const float r = cd2d_tap(grid, x, y, W);
        __builtin_nontemporal_store(r, &out[n - 1]);
    }
}

extern "C" void kernel_launch(void* const* d_in, const int* in_sizes, int n_in,
                              void* d_out, int out_size, void* d_ws, size_t ws_size,
                              hipStream_t stream) {
    // setup_inputs order: coords (N,2) int32, feats (N,1) f32, H int, W int
    const v4i*  coords2 = (const v4i*)d_in[0];
    const v2f*  feats2  = (const v2f*)d_in[1];
    const int*  pH      = (const int*)d_in[2];
    const int*  pW      = (const int*)d_in[3];
    float*      grid    = (float*)d_ws;               // H*W floats (64MB)
    float*      out     = (float*)d_out;

    const int n = in_sizes[1];  // N_POINTS (feats flat count)

    // Zero-fill: persistent grid-stride (H*W read on-device; capture-safe).
    cd2d_zero_kernel<<<3072, 256, 0, stream>>>(grid, pH, pW);

    // Pair-vectorized point kernels (+1 lane for odd tail).
    const int lanes  = (n >> 1) + 1;
    const int blocks = (lanes + 255) / 256;
    cd2d_scatter_kernel<<<blocks, 256, 0, stream>>>(coords2, feats2, grid, pW, n);
    cd2d_gather_kernel <<<blocks, 256, 0, stream>>>(coords2, grid, out, pW, n);
}